// PanopticSegmentor_22127671509696
// MI455X (gfx1250) — compile-verified
//
#include <hip/hip_runtime.h>
#include <hip/hip_bf16.h>

typedef __attribute__((ext_vector_type(2))) float v2f;
typedef __attribute__((ext_vector_type(8))) float v8f;

#define NUM_CLASSES 20
#define KPEAKS 256
#define NTILES (KPEAKS / 16)
#define PEAK_NMS 0.3f
#define OFFSET_NMS 25.0f
#define STRIDE_F 8.0f
#define FAR_SENT 1e8f
#define HM_H 64
#define HM_W 64
#define IMG 512
#define NPIX (IMG * IMG)
#define CAP 16384

// ---- workspace layout (bytes) ----
#define WS_COUNTER 0
#define WS_CAND 64
#define WS_BX (WS_CAND + CAP * 8)
#define WS_BY (WS_BX + KPEAKS * 4)
#define WS_BN (WS_BY + KPEAKS * 4)

__device__ __forceinline__ unsigned ordered_u32(float f) {
    unsigned b = __float_as_uint(f);
    return (b & 0x80000000u) ? ~b : (b | 0x80000000u);
}
__device__ __forceinline__ float unordered_f32(unsigned o) {
    unsigned b = (o & 0x80000000u) ? (o & 0x7FFFFFFFu) : ~o;
    return __uint_as_float(b);
}

// ---------- stage 0: zero the candidate counter ----------
__global__ void k_init(unsigned* counter) {
    if (threadIdx.x == 0 && blockIdx.x == 0) *counter = 0u;
}

// ---------- stage 1: 3x3 NMS peak candidates ----------
__global__ void k_peaks(const float* __restrict__ hmap, unsigned* counter,
                        unsigned long long* cand) {
    int i = blockIdx.x * blockDim.x + threadIdx.x;
    if (i >= HM_H * HM_W * NUM_CLASSES) return;
    int c = i % NUM_CLASSES;
    int x = (i / NUM_CLASSES) % HM_W;
    int y = i / (NUM_CLASSES * HM_W);
    float v = hmap[i];
    if (!(v > PEAK_NMS)) return;
    int y0 = y > 0 ? y - 1 : 0, y1 = y < HM_H - 1 ? y + 1 : HM_H - 1;
    int x0 = x > 0 ? x - 1 : 0, x1 = x < HM_W - 1 ? x + 1 : HM_W - 1;
    float mx = -__builtin_inff();
    for (int yy = y0; yy <= y1; ++yy)
        for (int xx = x0; xx <= x1; ++xx)
            mx = fmaxf(mx, hmap[(yy * HM_W + xx) * NUM_CLASSES + c]);
    if (v == mx) {
        unsigned pos = atomicAdd(counter, 1u);
        if (pos < CAP) {
            unsigned long long key =
                ((unsigned long long)ordered_u32(v) << 32) |
                (unsigned long long)(0xFFFFFFFFu - (unsigned)i);
            cand[pos] = key;
        }
    }
}

// ---------- stage 2: bitonic top-K sort + refine (single block, 1024 thr) ----------
__global__ void k_sort_refine(const float* __restrict__ rreg, unsigned* counter,
                              unsigned long long* cand, float* bx, float* by,
                              float* bn, float* out_cent, float* out_types,
                              float* out_scores, float* out_validpeak) {
    const int tid = threadIdx.x;
    const int nt = blockDim.x;
    unsigned count = *counter;
    if (count > CAP) count = CAP;
    for (int i = tid; i < CAP; i += nt)
        if ((unsigned)i >= count) cand[i] = 0ull;
    __syncthreads();
    // descending bitonic sort of packed (score, ~idx) keys
    for (unsigned k = 2; k <= CAP; k <<= 1) {
        for (unsigned j = k >> 1; j > 0; j >>= 1) {
            for (unsigned i = tid; i < CAP; i += nt) {
                unsigned ixj = i ^ j;
                if (ixj > i) {
                    unsigned long long a = cand[i], b = cand[ixj];
                    bool down = ((i & k) == 0);
                    if (down ? (a < b) : (a > b)) { cand[i] = b; cand[ixj] = a; }
                }
            }
            __syncthreads();
        }
    }
    if (tid < KPEAKS) {
        float score, cx, cy, cls;
        bool valid;
        if ((unsigned)tid < count) {
            unsigned long long key = cand[tid];
            score = unordered_f32((unsigned)(key >> 32));
            unsigned idx = 0xFFFFFFFFu - (unsigned)key;
            unsigned py = idx / (HM_W * NUM_CLASSES);
            unsigned rem = idx % (HM_W * NUM_CLASSES);
            unsigned px = rem / NUM_CLASSES;
            cls = (float)(rem % NUM_CLASSES);
            float r0 = rreg[(py * HM_W + px) * 2 + 0];
            float r1 = rreg[(py * HM_W + px) * 2 + 1];
            cx = ((float)px + r1) * STRIDE_F;  // centroids = cyx[:, ::-1] * stride
            cy = ((float)py + r0) * STRIDE_F;
            valid = score > PEAK_NMS;
        } else {
            score = -__builtin_inff(); cx = 0.f; cy = 0.f; cls = 0.f; valid = false;
        }
        out_cent[tid * 2 + 0] = cx;
        out_cent[tid * 2 + 1] = cy;
        out_types[tid] = cls;
        out_scores[tid] = score;
        out_validpeak[tid] = valid ? 1.f : 0.f;
        float cxp = valid ? cx : FAR_SENT;
        float cyp = valid ? cy : FAR_SENT;
        // lifted-distance B columns: D = ax*(-2cx) + ay*(-2cy) + |a|^2*1 + 1*|c|^2
        bx[tid] = -2.f * cxp;
        by[tid] = -2.f * cyp;
        bn[tid] = cxp * cxp + cyp * cyp;
    }
}

// ---------- stage 3: WMMA cdist + argmin + non_bg + outputs ----------
__global__ __launch_bounds__(256) void k_affil(
    const float* __restrict__ iseg, const float* __restrict__ sseg,
    const float* __restrict__ bx, const float* __restrict__ by,
    const float* __restrict__ bn, float* __restrict__ out_coords,
    float* __restrict__ out_aff, float* __restrict__ out_valid,
    float* __restrict__ out_oscore) {
    // B operand pre-packed in exact WMMA per-lane layout:
    // sB[t*32 + lane] = lane<16 ? (-2cx[n], -2cy[n]) : (1, |c_n|^2), n = t*16 + lane%16
    __shared__ v2f sB[NTILES * 32];
    __shared__ float sMin[8 * 16];
    const int tid = threadIdx.x;
    for (int i = tid; i < NTILES * 32; i += 256) {
        int l = i & 31;
        int n = ((i >> 5) << 4) + (l & 15);
        v2f v;
        if (l < 16) { v.x = bx[n]; v.y = by[n]; }
        else        { v.x = 1.0f;  v.y = bn[n]; }
        sB[i] = v;
    }
    __syncthreads();

    const int wave = tid >> 5;
    const int lane = tid & 31;
    const int r = lane & 15;      // A row within half / B column residue
    const int half = lane >> 4;   // 0: K={0,1}, 1: K={2,3}
    const int m0 = (blockIdx.x * 8 + wave) * 16;
    const int p = m0 + r;
    const int px = p & (IMG - 1);
    const int py = p >> 9;

    float oy = iseg[p * 2 + 0];   // iseg[..., ::-1]: ch1 -> x, ch0 -> y
    float ox = iseg[p * 2 + 1];
    float ax = (float)px + ox;
    float ay = (float)py + oy;
    float an = ax * ax + ay * ay;

    v2f A;                        // lanes 0-15: (ax, ay); lanes 16-31: (|a|^2, 1)
    A.x = half ? an : ax;
    A.y = half ? 1.0f : ay;

    // hoist all B tiles into registers: 16 ds_load_b64, one wait, then WMMA chain
    v2f Breg[NTILES];
#pragma unroll
    for (int t = 0; t < NTILES; ++t) Breg[t] = sB[t * 32 + lane];

    float minv[8];
    int mini[8];
#pragma unroll
    for (int v = 0; v < 8; ++v) { minv[v] = __builtin_inff(); mini[v] = 0; }

#pragma unroll
    for (int t = 0; t < NTILES; ++t) {
        v8f acc = {};
        acc = __builtin_amdgcn_wmma_f32_16x16x4_f32(false, A, false, Breg[t],
                                                    (short)0, acc, false, false);
        int n = t * 16 + r;
#pragma unroll
        for (int v = 0; v < 8; ++v) {   // lane holds D[row = v + 8*half][col = n]
            float d = acc[v];
            if (d < minv[v]) { minv[v] = d; mini[v] = n; }
        }
    }
    // reduce (min, argmin) across the 16 lanes of each half; ties -> lower index
#pragma unroll
    for (int off = 1; off < 16; off <<= 1) {
#pragma unroll
        for (int v = 0; v < 8; ++v) {
            float ov = __shfl_xor(minv[v], off, 32);
            int oi = __shfl_xor(mini[v], off, 32);
            if (ov < minv[v] || (ov == minv[v] && oi < mini[v])) {
                minv[v] = ov; mini[v] = oi;
            }
        }
    }
    if (r == 0) {                 // lane 0 -> rows 0-7, lane 16 -> rows 8-15
        int base = m0 + half * 8;
#pragma unroll
        for (int v = 0; v < 8; ++v) {
            out_oscore[base + v] = minv[v];
            out_aff[base + v] = (float)mini[v];
            sMin[wave * 16 + half * 8 + v] = minv[v];
        }
    }
    // same-wave LDS producer/consumer: DS ops are in-order per wave
    if (lane < 16) {
        float sc = sMin[wave * 16 + r];
        const float* ss = sseg + (size_t)p * (NUM_CLASSES + 1);
        float s0 = ss[0];
        bool nonbg = false;
        for (int c = 1; c <= NUM_CLASSES; ++c) nonbg |= (ss[c] > s0);
        out_valid[p] = (nonbg && sc < OFFSET_NMS) ? 1.f : 0.f;
        out_coords[p * 2 + 0] = (float)px;
        out_coords[p * 2 + 1] = (float)py;
    }
}

extern "C" void kernel_launch(void* const* d_in, const int* in_sizes, int n_in,
                              void* d_out, int out_size, void* d_ws,
                              size_t ws_size, hipStream_t stream) {
    const float* hmap = (const float*)d_in[0];
    const float* rreg = (const float*)d_in[1];
    const float* iseg = (const float*)d_in[2];
    const float* sseg = (const float*)d_in[3];

    char* ws = (char*)d_ws;
    unsigned* counter = (unsigned*)(ws + WS_COUNTER);
    unsigned long long* cand = (unsigned long long*)(ws + WS_CAND);
    float* bx = (float*)(ws + WS_BX);
    float* by = (float*)(ws + WS_BY);
    float* bn = (float*)(ws + WS_BN);

    float* out = (float*)d_out;
    float* o_coords = out;                   // [262144,2]
    float* o_aff = out + 524288;             // [262144]
    float* o_valid = out + 786432;           // [262144]
    float* o_oscore = out + 1048576;         // [262144]
    float* o_cent = out + 1310720;           // [256,2]
    float* o_types = out + 1311232;          // [256]
    float* o_scores = out + 1311488;         // [256]
    float* o_vpeak = out + 1311744;          // [256]

    k_init<<<1, 1, 0, stream>>>(counter);
    k_peaks<<<(HM_H * HM_W * NUM_CLASSES + 255) / 256, 256, 0, stream>>>(
        hmap, counter, cand);
    k_sort_refine<<<1, 1024, 0, stream>>>(rreg, counter, cand, bx, by, bn,
                                          o_cent, o_types, o_scores, o_vpeak);
    k_affil<<<NPIX / 128, 256, 0, stream>>>(iseg, sseg, bx, by, bn, o_coords,
                                            o_aff, o_valid, o_oscore);
}